// GeodynamicSpectralLayer_79233556677069
// MI455X (gfx1250) — compile-verified
//
#include <hip/hip_runtime.h>

#define NN 64
#define BATCH 4096
#define STEPS 8
#define MSTRIDE 66                 // padded LDS row stride (floats) for 64x64 matrices
#define MSLOT (NN * MSTRIDE)       // 4224 floats per matrix slot
#define LN_EPS 1e-5f

typedef float v2f __attribute__((ext_vector_type(2)));
typedef float v8f __attribute__((ext_vector_type(8)));

// ---------------------------------------------------------------------------
// 64x64 fp32 matmul on LDS using V_WMMA_F32_16X16X4_F32.
//   dst = (Cacc ? Cacc : 0) + op(A) * op(B),   op = optional transpose
// 256 threads = 8 wave32. Wave w computes tile row (w>>1), tile cols
// {(w&1)*2, (w&1)*2+1}: the A fragment is shared between two interleaved
// WMMA accumulator chains (c0,c1) -> half the A LDS loads, two independent
// dependency chains for the matrix pipe.
// Fragment layouts per CDNA5 ISA 7.12.2 (fp32, 16x16x4):
//   A 16x4 : lanes 0-15 -> K={0,1}, lanes 16-31 -> K={2,3}, M = lane&15
//   B 4x16 : lanes 0-15 -> K={0,1}, lanes 16-31 -> K={2,3}, N = lane&15
//   C 16x16: VGPR r -> row r (+8 for lanes 16-31), col = lane&15
// ---------------------------------------------------------------------------
__device__ __forceinline__ void mm64(float* __restrict__ dst,
                                     const float* __restrict__ A, bool tA,
                                     const float* __restrict__ Bm, bool tB,
                                     const float* __restrict__ Cacc, int tid)
{
    const int wave  = tid >> 5;
    const int lane  = tid & 31;
    const int mn    = lane & 15;
    const int khalf = (lane >> 4) << 1;    // 0 or 2
    const int cm0   = (lane >> 4) << 3;    // C row offset 0 or 8

    const int ti    = wave >> 1;           // tile row 0..3
    const int tj0   = (wave & 1) << 1;     // first tile col: 0 or 2
    const int rowA  = ti * 16 + mn;
    const int colB0 = tj0 * 16 + mn;       // second tile at colB0 + 16

    v8f c0 = {}, c1 = {};
    if (Cacc) {
        #pragma unroll
        for (int r = 0; r < 8; ++r) {
            c0[r] = Cacc[(ti * 16 + cm0 + r) * MSTRIDE + colB0];
            c1[r] = Cacc[(ti * 16 + cm0 + r) * MSTRIDE + colB0 + 16];
        }
    }

    #pragma unroll
    for (int kc = 0; kc < 16; ++kc) {
        const int k0 = kc * 4 + khalf;
        v2f a, b0, b1;
        if (!tA) {
            a.x = A[rowA * MSTRIDE + k0];
            a.y = A[rowA * MSTRIDE + k0 + 1];
        } else {
            a.x = A[k0 * MSTRIDE + rowA];
            a.y = A[(k0 + 1) * MSTRIDE + rowA];
        }
        if (!tB) {
            b0.x = Bm[k0 * MSTRIDE + colB0];
            b0.y = Bm[(k0 + 1) * MSTRIDE + colB0];
            b1.x = Bm[k0 * MSTRIDE + colB0 + 16];
            b1.y = Bm[(k0 + 1) * MSTRIDE + colB0 + 16];
        } else {
            b0.x = Bm[colB0 * MSTRIDE + k0];
            b0.y = Bm[colB0 * MSTRIDE + k0 + 1];
            b1.x = Bm[(colB0 + 16) * MSTRIDE + k0];
            b1.y = Bm[(colB0 + 16) * MSTRIDE + k0 + 1];
        }
        c0 = __builtin_amdgcn_wmma_f32_16x16x4_f32(false, a, false, b0, (short)0, c0, false, false);
        c1 = __builtin_amdgcn_wmma_f32_16x16x4_f32(false, a, false, b1, (short)0, c1, false, false);
    }

    #pragma unroll
    for (int r = 0; r < 8; ++r) {
        dst[(ti * 16 + cm0 + r) * MSTRIDE + colB0]      = c0[r];
        dst[(ti * 16 + cm0 + r) * MSTRIDE + colB0 + 16] = c1[r];
    }
}

// dst = s * (src - src^T)   (dst != src)
__device__ __forceinline__ void skewcomb(float* __restrict__ dst,
                                         const float* __restrict__ src,
                                         float s, int tid)
{
    #pragma unroll
    for (int i = 0; i < 16; ++i) {
        int e = tid + i * 256;
        int r = e >> 6, c = e & 63;
        dst[r * MSTRIDE + c] = s * (src[r * MSTRIDE + c] - src[c * MSTRIDE + r]);
    }
}

__device__ __forceinline__ void negate64(float* __restrict__ M, int tid)
{
    #pragma unroll
    for (int i = 0; i < 16; ++i) {
        int e = tid + i * 256;
        int r = e >> 6, c = e & 63;
        M[r * MSTRIDE + c] = -M[r * MSTRIDE + c];
    }
}

// ---------------------------------------------------------------------------
// One workgroup per batch element. 256 threads (8 wave32).
// ---------------------------------------------------------------------------
__global__ void __launch_bounds__(256)
geo_spectral_kernel(const float* __restrict__ x,  const float* __restrict__ U0,
                    const float* __restrict__ W1, const float* __restrict__ b1,
                    const float* __restrict__ lng,const float* __restrict__ lnb,
                    const float* __restrict__ W2, const float* __restrict__ b2,
                    const float* __restrict__ W3, const float* __restrict__ b3,
                    float* __restrict__ y)
{
    __shared__ float L[6 * MSLOT];           // ~99 KB of 320 KB WGP LDS
    __shared__ float xv[NN], hv[NN], hv2[NN];
    __shared__ float rowbuf[2][128];         // GJ: pivot row broadcast (double-buffered)
    __shared__ float fkbuf[2][NN];           // GJ: pivot column broadcast (double-buffered)

    const int tid  = threadIdx.x;
    const int bidx = blockIdx.x;

    float* Um  = L;                          // persistent: U
    float* Om  = L + 1 * MSLOT;              // persistent: Omega (skew)
    float* Gm  = L + 2 * MSLOT;              // persistent: G
    float* Vel = L + 3 * MSLOT;              // scratch: vel / v_corr
    float* Tm  = L + 4 * MSLOT;              // scratch: temporaries
    float* Wm  = L + 5 * MSLOT;              // scratch: w / comm / w_corr / X

    // ---- load x and U0 ----
    if (tid < NN) xv[tid] = x[bidx * NN + tid];
    #pragma unroll
    for (int i = 0; i < 16; ++i) {
        int e = tid + i * 256;
        int r = e >> 6, c = e & 63;
        Um[r * MSTRIDE + c] = U0[e];
    }
    __syncthreads();

    // ---- controller: Linear -> LayerNorm -> tanh ----
    if (tid < NN) {
        float a = b1[tid];
        for (int j = 0; j < NN; ++j) a += W1[tid * NN + j] * xv[j];
        hv[tid] = a;
    }
    __syncthreads();
    if (tid < NN) {
        float s = 0.f, s2 = 0.f;
        for (int j = 0; j < NN; ++j) { float v = hv[j]; s += v; s2 += v * v; }
        float mu  = s  * (1.0f / NN);
        float var = s2 * (1.0f / NN) - mu * mu;
        float t = (hv[tid] - mu) * rsqrtf(var + LN_EPS) * lng[tid] + lnb[tid];
        hv2[tid] = tanhf(t);
    }
    __syncthreads();
    // ---- Linear -> exact GELU ----
    if (tid < NN) {
        float a = b2[tid];
        for (int j = 0; j < NN; ++j) a += W2[tid * NN + j] * hv2[j];
        hv[tid] = 0.5f * a * (1.0f + erff(a * 0.70710678118654752f));
    }
    __syncthreads();

    // ---- p = h @ W3^T + b3 : A -> Tm, G -> Gm (wave-per-output, coalesced) ----
    {
        const int wave = tid >> 5, lane = tid & 31;
        const float hx = hv[lane], hy = hv[lane + 32];
        for (int i = 0; i < 1024; ++i) {
            const int o = wave * 1024 + i;
            const float* row = W3 + (size_t)o * NN;
            float part = row[lane] * hx + row[lane + 32] * hy;
            part += __shfl_xor(part, 16);
            part += __shfl_xor(part, 8);
            part += __shfl_xor(part, 4);
            part += __shfl_xor(part, 2);
            part += __shfl_xor(part, 1);
            if (lane == 0) {
                float v = part + b3[o];
                if (o < NN * NN) {
                    int r = o >> 6, c = o & 63;
                    Tm[r * MSTRIDE + c] = v;             // A
                } else {
                    int q = o - NN * NN;
                    int r = q >> 6, c = q & 63;
                    Gm[r * MSTRIDE + c] = v;             // G
                }
            }
        }
    }
    __syncthreads();
    skewcomb(Om, Tm, 0.5f, tid);             // Omega = 0.5*(A - A^T)
    __syncthreads();

    // ---- Lie-group integrator, 8 Cayley steps ----
    const float dt = 1.0f / STEPS;
    const float hh = 0.5f * dt;
    const int   rr = tid & 63;               // GJ: owned row
    const int   cg = tid >> 6;               // GJ: owned column group (32 cols)
    const int   cbase = cg * 32;

    #pragma unroll 1
    for (int st = 0; st < STEPS; ++st) {
        mm64(Tm, Um, true,  Gm, false, nullptr, tid);  __syncthreads(); // Tm = U^T G
        negate64(Tm, tid);                             __syncthreads(); // Tm = -U^T G
        mm64(Vel, Um, false, Om, false, Gm,  tid);     __syncthreads(); // vel = G + U*Om
        mm64(Vel, Um, false, Tm, false, Vel, tid);     __syncthreads(); // vel -= U*(U^T G)
        mm64(Tm, Vel, false, Um, true,  nullptr, tid); __syncthreads(); // P = vel*U^T
        skewcomb(Wm, Tm, 1.0f, tid);                   __syncthreads(); // w = P - P^T
        mm64(Tm, Om, false, Wm, false, nullptr, tid);  __syncthreads(); // P = Om*w
        skewcomb(Wm, Tm, hh, tid);                     __syncthreads(); // Wm = 0.5*dt*comm
        mm64(Vel, Um, false, Wm, false, Vel, tid);     __syncthreads(); // v_corr
        mm64(Tm, Vel, false, Um, true,  nullptr, tid); __syncthreads(); // P = v_corr*U^T
        skewcomb(Wm, Tm, 1.0f, tid);                   __syncthreads(); // w_corr

        // ---- Cayley: solve (I - h*W) X = (I + h*W), register-resident GJ ----
        // Thread (rr,cg) owns AUG[rr][cg*32 .. cg*32+31] in 32 VGPRs.
        // Double-buffered pivot broadcast -> single barrier per pivot.
        {
            float reg[32];
            if (cg < 2) {                    // left half: I - h*W
                #pragma unroll
                for (int c = 0; c < 32; ++c) {
                    int col = cbase + c;
                    reg[c] = (rr == col ? 1.0f : 0.0f) - hh * Wm[rr * MSTRIDE + col];
                }
            } else {                         // right half: I + h*W
                #pragma unroll
                for (int c = 0; c < 32; ++c) {
                    int col = cbase - NN + c;
                    reg[c] = (rr == col ? 1.0f : 0.0f) + hh * Wm[rr * MSTRIDE + col];
                }
            }

            // prologue: publish pivot 0 into buffer 0
            if (rr == 0) {
                #pragma unroll
                for (int c = 0; c < 32; ++c) rowbuf[0][cbase + c] = reg[c];
            }
            if (cg == 0) fkbuf[0][rr] = reg[0];
            __syncthreads();

            // pivot loop: unrolled in blocks of 32 so register indices and
            // buffer parities are compile-time constants
            #pragma unroll 1
            for (int k2 = 0; k2 < 2; ++k2) {
                #pragma unroll
                for (int ku = 0; ku < 32; ++ku) {
                    const int k   = k2 * 32 + ku;
                    const int par = ku & 1;              // == k & 1
                    const float inv = 1.0f / rowbuf[par][k];
                    if (rr == k) {                       // normalize pivot row
                        #pragma unroll
                        for (int c = 0; c < 32; ++c) reg[c] *= inv;
                    } else {                             // eliminate
                        const float f = fkbuf[par][rr] * inv;
                        #pragma unroll
                        for (int c = 0; c < 32; ++c) reg[c] -= f * rowbuf[par][cbase + c];
                    }
                    // publish pivot k+1 into the other buffer (values already
                    // reflect eliminations 0..k in this thread's registers)
                    const int kn = k + 1;
                    if (kn < NN) {
                        const int npar = par ^ 1;
                        if (rr == kn) {
                            #pragma unroll
                            for (int c = 0; c < 32; ++c) rowbuf[npar][cbase + c] = reg[c];
                        }
                        if (cg == (kn >> 5)) fkbuf[npar][rr] = reg[(ku + 1) & 31];
                    }
                    __syncthreads();
                }
            }

            // X = right half -> Wm
            if (cg >= 2) {
                #pragma unroll
                for (int c = 0; c < 32; ++c)
                    Wm[rr * MSTRIDE + (cbase - NN) + c] = reg[c];
            }
        }
        __syncthreads();

        mm64(Tm, Wm, false, Um, false, nullptr, tid);  __syncthreads(); // U_new = X*U
        #pragma unroll
        for (int i = 0; i < 16; ++i) {
            int e = tid + i * 256;
            int r = e >> 6, c = e & 63;
            Um[r * MSTRIDE + c] = Tm[r * MSTRIDE + c];
        }
        __syncthreads();
    }

    // ---- y = U x ----
    if (tid < NN) {
        float a = 0.f;
        for (int j = 0; j < NN; ++j) a += Um[tid * MSTRIDE + j] * xv[j];
        y[bidx * NN + tid] = a;
    }
}

extern "C" void kernel_launch(void* const* d_in, const int* in_sizes, int n_in,
                              void* d_out, int out_size, void* d_ws, size_t ws_size,
                              hipStream_t stream)
{
    const float* x   = (const float*)d_in[0];
    const float* U0  = (const float*)d_in[1];
    const float* W1  = (const float*)d_in[2];
    const float* b1  = (const float*)d_in[3];
    const float* lng = (const float*)d_in[4];
    const float* lnb = (const float*)d_in[5];
    const float* W2  = (const float*)d_in[6];
    const float* b2  = (const float*)d_in[7];
    const float* W3  = (const float*)d_in[8];
    const float* b3  = (const float*)d_in[9];

    geo_spectral_kernel<<<BATCH, 256, 0, stream>>>(
        x, U0, W1, b1, lng, lnb, W2, b2, W3, b3, (float*)d_out);
}